// PointsMasksMatcher_18305150615903
// MI455X (gfx1250) — compile-verified
//
#include <hip/hip_runtime.h>
#include <math.h>

#define B_ 4
#define P_ 16384
#define G_ 512
#define H_ 512
#define W_ 512

typedef float v2f __attribute__((ext_vector_type(2)));
typedef float v8f __attribute__((ext_vector_type(8)));
typedef int   v4i __attribute__((ext_vector_type(4)));

// key bias: BIG > max d^2 = 2*512^2 = 2^19. Inside keys = d2 (exact),
// outside keys = d2 + BIG. has_inside <=> minKey < 1.5*2^19 (margin covers
// tiny negative d2 from the |p|^2+|g|^2-2p.g expansion).
#define BIGF 1048576.0f   /* 2^20 */
#define THRF 786432.0f    /* 1.5 * 2^19 */

// ---------------------------------------------------------------------------
// Pass 1: gather label at each rounded/clamped pred-point location.
// labels[b*P + p] = label_maps[b, clamp(round(y)), clamp(round(x))]
// ---------------------------------------------------------------------------
__global__ __launch_bounds__(256)
void pm_point_labels_kernel(const float* __restrict__ pred,
                            const int*   __restrict__ lmap,
                            int*         __restrict__ labels) {
  int idx = blockIdx.x * blockDim.x + threadIdx.x;   // 0 .. B*P-1
  if (idx >= B_ * P_) return;
  int b = idx / P_;
  float px = pred[2 * idx + 0];
  float py = pred[2 * idx + 1];
  // jnp.round = round-half-to-even -> rintf (v_rndne_f32)
  int x = (int)rintf(px); x = min(max(x, 0), W_ - 1);
  int y = (int)rintf(py); y = min(max(y, 0), H_ - 1);
  labels[idx] = lmap[b * (H_ * W_) + y * W_ + x];
}

// ---------------------------------------------------------------------------
// Pass 2: per (batch, 16-gt group) block. Each wave computes 16x16 tiles of
// the squared-distance matrix with V_WMMA_F32_16X16X4_F32, folding the
// rank-1 terms into the K=4 slots:
//   A[m] = (-2px, -2py, |p|^2, 1)   B[n] = (gx, gy, 1, |g|^2)
//   => C[m][n] = D^2[m][n]
// Single-track biased-key argmin over P replaces the inside/all dual track.
// ---------------------------------------------------------------------------
__global__ __launch_bounds__(256)
void pm_match_kernel(const float* __restrict__ pred,
                     const float* __restrict__ gt,
                     const int*   __restrict__ labels,
                     float*       __restrict__ out) {
  const int groupsPerBatch = G_ / 16;                // 32
  int b     = blockIdx.x / groupsPerBatch;
  int gbase = (blockIdx.x % groupsPerBatch) * 16;

  int tid  = threadIdx.x;
  int wave = tid >> 5;          // 0..7 (8 waves per block)
  int lane = tid & 31;          // wave32
  int half = lane >> 4;         // 0: K0/K1 slots, 1: K2/K3 slots
  int n    = lane & 15;         // this lane's gt column within the group
  int g    = gbase + n;         // 0-based gt index

  // ---- B matrix: constant over the P loop --------------------------------
  float gx  = gt[((size_t)b * G_ + g) * 2 + 0];
  float gy  = gt[((size_t)b * G_ + g) * 2 + 1];
  float gn2 = gx * gx + gy * gy;
  v2f bm;
  bm.x = half ? 1.0f : gx;      // K2 = 1        | K0 = gx
  bm.y = half ? gn2  : gy;      // K3 = |g|^2    | K1 = gy

  float bestK = __builtin_inff();
  int   bestM = 0;
  const int matchLab = g + 1;   // label id for instance g

  const float* predb = pred   + (size_t)b * P_ * 2;
  const int*   labb  = labels + (size_t)b * P_;

  // ---- sweep P in 16-row chunks, 8 waves interleaved ---------------------
  for (int chunk = wave; chunk < P_ / 16; chunk += 8) {
    int base = chunk * 16;
    int m0   = base + n;                      // row this lane loads
    float px = predb[2 * m0 + 0];
    float py = predb[2 * m0 + 1];
    float pn2 = px * px + py * py;

    // prefetch next chunk for this wave (stride = 8 chunks = 128 rows);
    // lowers to global_prefetch_b8 (speculative, safe past end)
    __builtin_prefetch(predb + 2 * (m0 + 128), 0, 3);
    __builtin_prefetch(labb + (base + 8 * half + 128), 0, 3);

    v2f am;
    am.x = half ? pn2  : -2.0f * px;          // K2 = |p|^2 | K0 = -2px
    am.y = half ? 1.0f : -2.0f * py;          // K3 = 1     | K1 = -2py

    v8f c = {0.f, 0.f, 0.f, 0.f, 0.f, 0.f, 0.f, 0.f};
    // D = A x B + C : full 16x16 squared-distance tile in one instruction
    c = __builtin_amdgcn_wmma_f32_16x16x4_f32(
            /*neg_a=*/false, am, /*neg_b=*/false, bm,
            /*c_mod=*/(short)0, c, /*reuse_a=*/false, /*reuse_b=*/false);

    // C layout: this lane holds column n, rows m = base + 8*half + v
    int rbase = base + 8 * half;
    const v4i* lp = (const v4i*)(labb + rbase);   // 32B-aligned (rbase % 8 == 0)
    v4i lab0 = lp[0];
    v4i lab1 = lp[1];

#pragma unroll
    for (int v = 0; v < 8; ++v) {
      float d2  = c[v];
      int   lab = (v < 4) ? lab0[v] : lab1[v - 4];
      float key = (lab == matchLab) ? d2 : (d2 + BIGF);
      int   m   = rbase + v;
      // rows ascend within this lane's sequence -> strict '<' keeps first idx
      if (key < bestK) { bestK = key; bestM = m; }
    }
  }

  // ---- merge the two lane-halves (same column n, disjoint row sets) ------
  {
    float oK = __shfl_xor(bestK, 16);
    int   oM = __shfl_xor(bestM, 16);
    if (oK < bestK || (oK == bestK && oM < bestM)) { bestK = oK; bestM = oM; }
  }

  // ---- cross-wave reduction through LDS ----------------------------------
  __shared__ float sK[8][16];
  __shared__ int   sM[8][16];
  if (lane < 16) { sK[wave][n] = bestK; sM[wave][n] = bestM; }
  __syncthreads();

  if (tid < 16) {
    float fK = sK[0][tid]; int iM = sM[0][tid];
#pragma unroll
    for (int w = 1; w < 8; ++w) {
      float v = sK[w][tid]; int i = sM[w][tid];
      if (v < fK || (v == fK && i < iM)) { fK = v; iM = i; }
    }
    bool  has  = fK < THRF;                    // has_inside[b, g]
    float d2   = has ? fK : (fK - BIGF);       // un-bias the outside track
    float cost = sqrtf(fmaxf(d2, 0.0f));       // monotone map back from D^2
    int   gg   = gbase + tid;

    size_t off = (size_t)b * G_ + gg;
    const size_t SEC = (size_t)B_ * G_;
    out[off]           = (float)iM;            // src_idx  [B,G]
    out[SEC + off]     = (float)gg;            // tgt_idx  [B,G] = arange(G)
    out[2 * SEC + off] = cost;                 // costs    [B,G]
  }
}

// ---------------------------------------------------------------------------
extern "C" void kernel_launch(void* const* d_in, const int* in_sizes, int n_in,
                              void* d_out, int out_size, void* d_ws, size_t ws_size,
                              hipStream_t stream) {
  const float* pred = (const float*)d_in[0];   // [B,P,2] f32
  const float* gt   = (const float*)d_in[1];   // [B,G,2] f32
  const int*   lmap = (const int*)  d_in[2];   // [B,H,W] i32
  float*       out  = (float*)d_out;           // [3, B, G] f32 (src, tgt, costs)
  int*         labels = (int*)d_ws;            // scratch: B*P ints = 256 KB

  const int threads = 256;
  const int blocks1 = (B_ * P_ + threads - 1) / threads;
  pm_point_labels_kernel<<<blocks1, threads, 0, stream>>>(pred, lmap, labels);

  const int blocks2 = B_ * (G_ / 16);          // 128 blocks x 8 waves
  pm_match_kernel<<<blocks2, threads, 0, stream>>>(pred, gt, labels, out);
}